// SpKBGCN_4329327034642
// MI455X (gfx1250) — compile-verified
//
#include <hip/hip_runtime.h>
#include <math.h>

typedef __attribute__((ext_vector_type(2))) float v2f;
typedef __attribute__((ext_vector_type(8))) float v8f;

#define N_TOT   50500
#define N_EDGE  800000
#define D_IN    100
#define D_HID   200
#define ALPHA   0.2f
#define EPS_N   1e-12f

// ---------------------------------------------------------------------------
// Dense GEMM C[N_TOT x 200] = A[N_TOT x K] @ B[K x 200] with
// V_WMMA_F32_16X16X4_F32.  One wave owns one 16x16 C tile.
//
// Per-lane operand layout (ISA 7.12.2, 32-bit A 16x4 / B 4x16):
//   lane = 16*half + l
//   A: a.x = A[m0+l][k0+2*half], a.y = A[m0+l][k0+2*half+1]  (consecutive -> b64)
//   B: b.x = B[k0+2*half][n0+l], b.y = B[k0+2*half+1][n0+l]  (row stride 200)
//   D vgpr r: C[m0 + r + 8*half][n0 + l]
//
// Out-of-range rows/cols are CLAMPED (not masked): a bogus A-row m only
// pollutes C-row m and a bogus B-col n only pollutes C-col n, and those are
// masked at the store.  This keeps all inner-loop loads unconditional (no
// exec-mask churn) and EXEC all-ones at every WMMA (wave-uniform exit only).
// Two accumulator chains (even/odd k-blocks) expose WMMA ILP.
// ---------------------------------------------------------------------------
__global__ void gcn_wmma_gemm_f32(const float* __restrict__ A,
                                  const float* __restrict__ B,
                                  float* __restrict__ C,
                                  int K) {
  const int wave = (blockIdx.x * blockDim.x + threadIdx.x) >> 5;
  const int lane = threadIdx.x & 31;
  const int ntiles = D_HID / 16 + 1;            // 13 (last tile half-covered)
  const int mtiles = (N_TOT + 15) >> 4;         // 3157
  const int mt = wave / ntiles;
  const int nt = wave - mt * ntiles;
  if (mt >= mtiles) return;                     // wave-uniform: EXEC stays full

  const int m0 = mt << 4;
  const int n0 = nt << 4;
  const int half = lane >> 4;
  const int l    = lane & 15;

  const bool mok = (m0 + l) < N_TOT;
  const bool nok = (n0 + l) < D_HID;
  const int mrow = mok ? (m0 + l) : (N_TOT - 1); // clamped (see note above)
  const int ncol = nok ? (n0 + l) : (D_HID - 1);

  const float* aPtr = A + (size_t)mrow * K + 2 * half;       // += 4 / k-block
  const float* bPtr = B + (size_t)(2 * half) * D_HID + ncol; // += 4*200 / k-block

  v8f acc0 = {};
  v8f acc1 = {};
  int k0 = 0;
  for (; k0 + 8 <= K; k0 += 8) {                // 2 independent WMMA chains
    v2f a0 = *(const v2f*)aPtr;
    v2f a1 = *(const v2f*)(aPtr + 4);
    v2f b0; b0.x = bPtr[0];           b0.y = bPtr[D_HID];
    v2f b1; b1.x = bPtr[4 * D_HID];   b1.y = bPtr[5 * D_HID];
    acc0 = __builtin_amdgcn_wmma_f32_16x16x4_f32(false, a0, false, b0,
                                                 (short)0, acc0, false, false);
    acc1 = __builtin_amdgcn_wmma_f32_16x16x4_f32(false, a1, false, b1,
                                                 (short)0, acc1, false, false);
    aPtr += 8;
    bPtr += 8 * D_HID;
  }
  for (; k0 < K; k0 += 4) {                     // K=100 tail (one k-block)
    v2f a0 = *(const v2f*)aPtr;
    v2f b0; b0.x = bPtr[0]; b0.y = bPtr[D_HID];
    acc0 = __builtin_amdgcn_wmma_f32_16x16x4_f32(false, a0, false, b0,
                                                 (short)0, acc0, false, false);
    aPtr += 4;
    bPtr += 4 * D_HID;
  }

  if (nok) {
#pragma unroll
    for (int r = 0; r < 8; ++r) {
      const int mr = m0 + r + 8 * half;
      if (mr < N_TOT) C[(size_t)mr * D_HID + ncol] = acc0[r] + acc1[r];
    }
  }
}

// ---------------------------------------------------------------------------
// Edge gather-scale-scatter: one wave per edge, lanes stride the 200 columns.
// agg[dst][c] += h[src][c] * val  via global_atomic_add_f32.
// Indices forced into SGPRs (wave-uniform) -> scalar row-base addressing.
// ---------------------------------------------------------------------------
__global__ void gcn_edge_scatter(const float* __restrict__ H,
                                 const int* __restrict__ src,
                                 const int* __restrict__ dst,
                                 const float* __restrict__ vals,
                                 float* __restrict__ Agg) {
  const int e    = (blockIdx.x * blockDim.x + threadIdx.x) >> 5;
  const int lane = threadIdx.x & 31;
  if (e >= N_EDGE) return;                      // wave-uniform
  const int   s = __builtin_amdgcn_readfirstlane(src[e]);
  const int   d = __builtin_amdgcn_readfirstlane(dst[e]);
  const float v = vals[e];
  const float* __restrict__ hrow = H + (size_t)s * D_HID;
  float* __restrict__ arow = Agg + (size_t)d * D_HID;
#pragma unroll
  for (int c = lane; c < D_HID; c += 32)
    atomicAdd(&arow[c], hrow[c] * v);
}

// ---------------------------------------------------------------------------
// Elementwise LeakyReLU in place.
// ---------------------------------------------------------------------------
__global__ void gcn_leaky(float* __restrict__ X, size_t n) {
  size_t i = (size_t)blockIdx.x * blockDim.x + threadIdx.x;
  if (i >= n) return;
  float x = X[i];
  X[i] = (x > 0.f) ? x : ALPHA * x;
}

// ---------------------------------------------------------------------------
// Fused LeakyReLU + row L2-normalize, one wave per row (wave32 shuffles).
// ---------------------------------------------------------------------------
__global__ void gcn_leaky_l2norm(float* __restrict__ X) {
  const int row  = (blockIdx.x * blockDim.x + threadIdx.x) >> 5;
  const int lane = threadIdx.x & 31;
  if (row >= N_TOT) return;
  float* __restrict__ x = X + (size_t)row * D_HID;

  float y[7];
  float ss = 0.f;
  int k = 0;
  for (int c = lane; c < D_HID; c += 32, ++k) {
    float v = x[c];
    v = (v > 0.f) ? v : ALPHA * v;
    y[k] = v;
    ss += v * v;
  }
#pragma unroll
  for (int off = 16; off > 0; off >>= 1)
    ss += __shfl_xor(ss, off, 32);
  const float inv = 1.f / fmaxf(sqrtf(ss), EPS_N);
  k = 0;
  for (int c = lane; c < D_HID; c += 32, ++k)
    x[c] = y[k] * inv;
}

// ---------------------------------------------------------------------------
// Launch: emb@W1 -> scatter -> leaky -> h1@W2 -> scatter -> leaky+norm
// ---------------------------------------------------------------------------
extern "C" void kernel_launch(void* const* d_in, const int* in_sizes, int n_in,
                              void* d_out, int out_size, void* d_ws, size_t ws_size,
                              hipStream_t stream) {
  const float* emb  = (const float*)d_in[0];   // [N_TOT, D_IN]
  const float* W1   = (const float*)d_in[1];   // [D_IN, D_HID]
  const float* W2   = (const float*)d_in[2];   // [D_HID, D_HID]
  const int*   eidx = (const int*)d_in[3];     // [2, E] (row0=src, row1=dst)
  const float* vals = (const float*)d_in[4];   // [E]
  float*       out  = (float*)d_out;           // [N_TOT, D_HID]

  const int* src = eidx;
  const int* dst = eidx + N_EDGE;

  const size_t bufElems = (size_t)N_TOT * D_HID;
  float* bufA = (float*)d_ws;                  // projection output
  float* bufB = bufA + bufElems;               // layer-1 aggregation / h1

  const int threads = 256;                     // 8 waves per block
  const int wavesPerBlk = threads / 32;

  const int mtiles = (N_TOT + 15) / 16;        // 3157
  const int ntiles = D_HID / 16 + 1;           // 13
  const int gemmWaves = mtiles * ntiles;
  const int gemmBlks  = (gemmWaves + wavesPerBlk - 1) / wavesPerBlk;

  const int scatBlks  = (N_EDGE + wavesPerBlk - 1) / wavesPerBlk;
  const int eltBlks   = (int)((bufElems + threads - 1) / threads);
  const int rowBlks   = (N_TOT + wavesPerBlk - 1) / wavesPerBlk;

  // ---- Layer 1 ----
  gcn_wmma_gemm_f32<<<gemmBlks, threads, 0, stream>>>(emb, W1, bufA, D_IN);
  hipMemsetAsync(bufB, 0, bufElems * sizeof(float), stream);
  gcn_edge_scatter<<<scatBlks, threads, 0, stream>>>(bufA, src, dst, vals, bufB);
  gcn_leaky<<<eltBlks, threads, 0, stream>>>(bufB, bufElems);

  // ---- Layer 2 ----
  gcn_wmma_gemm_f32<<<gemmBlks, threads, 0, stream>>>(bufB, W2, bufA, D_HID);
  hipMemsetAsync(out, 0, bufElems * sizeof(float), stream);
  gcn_edge_scatter<<<scatBlks, threads, 0, stream>>>(bufA, src, dst, vals, out);

  // ---- LeakyReLU + L2 normalize rows ----
  gcn_leaky_l2norm<<<rowBlks, threads, 0, stream>>>(out);
}